// Model_23880018165873
// MI455X (gfx1250) — compile-verified
//
#include <hip/hip_runtime.h>

// MoE weighted scatter-add combine for MI455X (gfx1250).
//
// out[token_idx[r], :] += gates[r] * expert_hidden[r, :]
//   R = 131072 rows, D = 2048 fp32, T = 65536 tokens.
//
// Memory-bound: ~2 GiB total traffic -> ~90us at 23.3 TB/s. No matmul ->
// no WMMA (matrix units cannot help a bandwidth-bound scatter). Levers
// (all confirmed in gfx1250 disasm):
//   - global_load_b32 th:TH_LOAD_NT  : stream the single-use 1 GiB input
//     without polluting the 192 MB L2, which the atomic units need for the
//     512 MB output RMW working set.
//   - global_atomic_add_f32 SCOPE_DEV: non-returning HW f32 atomic (no CAS).
//   - s_load for block-uniform token/gate, scale_offset addressing.
//   - wave32-contiguous layout: each wave's load/atomic covers 128B spans.
// This round: batch the 4 loads into distinct registers so they issue as one
// s_clause with a single s_wait_loadcnt (prev codegen reused one VGPR and
// serialized load->wait->atomic x4).

#define D_MODEL 2048
#define SCATTER_THREADS 512   // 16 waves of 32; one block per row
#define ELEMS_PER_THREAD (D_MODEL / SCATTER_THREADS)   // 4

// ---------------------------------------------------------------------------
// Weighted scatter-add. One 512-thread block per row.
// ---------------------------------------------------------------------------
__global__ __launch_bounds__(SCATTER_THREADS)
void moe_combine_scatter_kernel(const float* __restrict__ hidden,
                                const int*   __restrict__ token_idx,
                                const float* __restrict__ gates,
                                float*       __restrict__ out) {
    const int row = blockIdx.x;
    const int tok = token_idx[row];            // uniform -> s_load
    const float g = gates[row];                // uniform -> s_load

    const float* __restrict__ src = hidden + (size_t)row * D_MODEL;
    float*       __restrict__ dst = out    + (size_t)tok * D_MODEL;

    const int d = threadIdx.x;

    // Issue all loads first (distinct registers -> one clause, one wait).
    float v[ELEMS_PER_THREAD];
#pragma unroll
    for (int k = 0; k < ELEMS_PER_THREAD; ++k) {
        v[k] = __builtin_nontemporal_load(src + d + k * SCATTER_THREADS);
    }

    // Then fire all non-returning HW atomics back-to-back.
#pragma unroll
    for (int k = 0; k < ELEMS_PER_THREAD; ++k) {
        unsafeAtomicAdd(dst + d + k * SCATTER_THREADS, v[k] * g);
    }
}

// ---------------------------------------------------------------------------
// Zero the output (harness poisons d_out; atomics accumulate, so every call
// must start from zeros to stay deterministic across graph replays).
// ---------------------------------------------------------------------------
__global__ __launch_bounds__(256)
void zero_f4_kernel(float4* __restrict__ out, int n4) {
    const int stride = gridDim.x * blockDim.x;
    float4 z;
    z.x = 0.0f; z.y = 0.0f; z.z = 0.0f; z.w = 0.0f;
    for (int i = blockIdx.x * blockDim.x + threadIdx.x; i < n4; i += stride) {
        out[i] = z;
    }
}

// ---------------------------------------------------------------------------
// Host-side launcher (graph-capture safe: only kernel launches on `stream`).
// Inputs (setup_inputs order):
//   d_in[0] expert_hidden : float32 [R * D]
//   d_in[1] token_idx     : int32   [R]
//   d_in[2] gates         : float32 [R]
//   d_in[3] num_tokens    : int32   [1]   (implied by out_size; unused)
// Output: float32 [T * D]
// ---------------------------------------------------------------------------
extern "C" void kernel_launch(void* const* d_in, const int* in_sizes, int n_in,
                              void* d_out, int out_size, void* d_ws, size_t ws_size,
                              hipStream_t stream) {
    (void)n_in; (void)d_ws; (void)ws_size;

    const float* hidden    = (const float*)d_in[0];
    const int*   token_idx = (const int*)  d_in[1];
    const float* gates     = (const float*)d_in[2];
    float*       out       = (float*)d_out;

    const int num_rows = in_sizes[1];          // 131072
    const int n4       = out_size / 4;         // float4 count (out_size % 4 == 0)

    // Phase 1: zero 512 MiB output with wide stores.
    zero_f4_kernel<<<8192, 256, 0, stream>>>((float4*)d_out, n4);

    // Phase 2: one block per row, atomic scatter-add.
    moe_combine_scatter_kernel<<<num_rows, SCATTER_THREADS, 0, stream>>>(
        hidden, token_idx, gates, out);
}